// CEPLoss_28741921145434
// MI455X (gfx1250) — compile-verified
//
#include <hip/hip_runtime.h>

// loss = -sum(d1 * log(d2 + 1e-5)) over 4096x4096 fp32.
// HBM-bound streaming reduce: 128 MiB in -> 4 B out. Floor ~5.8us @ 23.3 TB/s;
// on graph replay both inputs stay resident in the 192 MB L2 (RT loads), where
// the limit shifts to VALU -- so per-element work is trimmed to {add, v_log_f32, fma}
// by computing in log2 domain and applying the -ln2 factor once at the end:
//   loss = -ln2 * sum(d1 * log2(d2 + eps))
// Raw v_log_f32 (no denorm fixup) is safe: argument >= 1e-5 >> FLT_MIN.

#define EPSF 1e-5f
#define NEG_LN2F (-0.69314718055994530942f)

constexpr int N2      = 4096 * 4096;   // total elements
constexpr int VEC     = 4;             // float4 / global_load_b128
constexpr int NV      = N2 / VEC;      // 4,194,304 vec4 chunks
constexpr int BLOCKS  = 2048;          // partial-sum blocks (ws: 8 KiB)
constexpr int THREADS = 256;           // 8 waves/block (wave32)

typedef __attribute__((ext_vector_type(2))) float v2f;
typedef __attribute__((ext_vector_type(8))) float v8f;

// Wave32 sum via V_WMMA_F32_16X16X4_F32 (matrix pipe, co-executes with VALU).
// A (16x4 f32, 2 VGPRs): lanes 0-15 give A[m,0]=x_m, lanes 16-31 give A[m,2]=x_{m+16};
// K=1,3 slots are zero. B = all ones (layout-independent). C = 0.
// => D[m,n] = x_m + x_{m+16} for all n.
// Summing a lane's 8 D VGPRs yields the half-sum for its 8 rows (lanes 0-15: rows 0-7,
// lanes 16-31: rows 8-15); one xor-16 shuffle merges the halves.
// Requires EXEC == all ones (call only from fully-converged waves).
__device__ __forceinline__ float wave_sum_wmma(float x) {
    v2f a; a[0] = x;    a[1] = 0.0f;
    v2f b; b[0] = 1.0f; b[1] = 1.0f;
    v8f c = {};
    v8f d = __builtin_amdgcn_wmma_f32_16x16x4_f32(
        /*neg_a=*/false, a, /*neg_b=*/false, b,
        /*c_mod=*/(short)0, c, /*reuse_a=*/false, /*reuse_b=*/false);
    float s = ((d[0] + d[1]) + (d[2] + d[3])) + ((d[4] + d[5]) + (d[6] + d[7]));
    s += __shfl_xor(s, 16, 32);
    return s;                     // full 32-lane sum, broadcast to all lanes
}

__global__ __launch_bounds__(THREADS)
void cep_partial_kernel(const float4* __restrict__ a4,
                        const float4* __restrict__ b4,
                        float* __restrict__ partials) {
    const int tid    = blockIdx.x * blockDim.x + threadIdx.x;
    const int stride = gridDim.x * blockDim.x;

    float acc = 0.0f;
    // 4,194,304 / (2048*256) = 8 full b128-pair iterations per thread, no tail.
    for (int i = tid; i < NV; i += stride) {
        float4 a = a4[i];
        float4 b = b4[i];
        // raw v_log_f32 (log2); argument in [1e-5, ~1.0001] -- no denorm fixup needed
        acc = fmaf(a.x, __builtin_amdgcn_logf(b.x + EPSF), acc);
        acc = fmaf(a.y, __builtin_amdgcn_logf(b.y + EPSF), acc);
        acc = fmaf(a.z, __builtin_amdgcn_logf(b.z + EPSF), acc);
        acc = fmaf(a.w, __builtin_amdgcn_logf(b.w + EPSF), acc);
    }

    // Wave reduce on the matrix pipe (all lanes converged here).
    float s = wave_sum_wmma(acc);

    __shared__ float wsum[THREADS / 32];
    const int lane = threadIdx.x & 31;
    const int wave = threadIdx.x >> 5;
    if (lane == 0) wsum[wave] = s;
    __syncthreads();
    if (threadIdx.x == 0) {
        float t = 0.0f;
        #pragma unroll
        for (int w = 0; w < THREADS / 32; ++w) t += wsum[w];
        partials[blockIdx.x] = t;   // fixed slot -> deterministic
    }
}

__global__ __launch_bounds__(1024)
void cep_final_kernel(const float* __restrict__ partials,
                      float* __restrict__ out) {
    // 1024 threads reduce BLOCKS=2048 partials deterministically.
    float acc = partials[threadIdx.x] + partials[threadIdx.x + 1024];

    float s = wave_sum_wmma(acc);   // 32 waves, each fully active

    __shared__ float wsum[32];
    const int lane = threadIdx.x & 31;
    const int wave = threadIdx.x >> 5;
    if (lane == 0) wsum[wave] = s;
    __syncthreads();

    if (threadIdx.x < 32) {         // whole wave 0 -> EXEC all ones inside
        float v = wsum[threadIdx.x];
        float t = wave_sum_wmma(v);
        // sum is in log2 domain; fold in -ln2 once (loss = -ln2 * sum)
        if (threadIdx.x == 0) out[0] = t * NEG_LN2F;
    }
}

extern "C" void kernel_launch(void* const* d_in, const int* in_sizes, int n_in,
                              void* d_out, int out_size, void* d_ws, size_t ws_size,
                              hipStream_t stream) {
    const float* d1 = (const float*)d_in[0];   // distribution1, fp32 4096x4096
    const float* d2 = (const float*)d_in[1];   // distribution2, fp32 4096x4096
    float* out      = (float*)d_out;           // 1 fp32 element
    float* partials = (float*)d_ws;            // BLOCKS floats = 8 KiB scratch

    cep_partial_kernel<<<BLOCKS, THREADS, 0, stream>>>(
        (const float4*)d1, (const float4*)d2, partials);
    cep_final_kernel<<<1, 1024, 0, stream>>>(partials, out);
}